// ConstrainedSparsemax_47785806135581
// MI455X (gfx1250) — compile-verified
//
#include <hip/hip_runtime.h>
#include <stdint.h>

typedef __attribute__((ext_vector_type(4))) float v4f;
typedef __attribute__((ext_vector_type(4))) int   v4i;

#define KDIM            1024
#define CHUNKS          8      // 8 chunks * (32 lanes * 4 floats) = 1024 elems/row
#define WAVES_PER_BLOCK 4
#define LANES           32
#define BISECT_ITERS    12
#define NEWTON_ITERS    6

__device__ __forceinline__ float wred_sum(float x) {
#pragma unroll
  for (int off = 16; off > 0; off >>= 1) x += __shfl_xor(x, off, 32);
  return x;
}
__device__ __forceinline__ float wred_max(float x) {
#pragma unroll
  for (int off = 16; off > 0; off >>= 1) x = fmaxf(x, __shfl_xor(x, off, 32));
  return x;
}
__device__ __forceinline__ float wred_min(float x) {
#pragma unroll
  for (int off = 16; off > 0; off >>= 1) x = fminf(x, __shfl_xor(x, off, 32));
  return x;
}

__global__ __launch_bounds__(WAVES_PER_BLOCK * LANES)
void ConstrainedSparsemax_kernel(const float* __restrict__ z_g,
                                 const float* __restrict__ u_g,
                                 float* __restrict__ p_out,
                                 int*   __restrict__ r_out,
                                 float* __restrict__ tau_out,
                                 float* __restrict__ val_out,
                                 int B) {
  __shared__ __align__(16) float s_z[WAVES_PER_BLOCK][KDIM];
  __shared__ __align__(16) float s_u[WAVES_PER_BLOCK][KDIM];

  const int lane = threadIdx.x & (LANES - 1);
  const int wave = threadIdx.x >> 5;
  const int row  = blockIdx.x * WAVES_PER_BLOCK + wave;
  if (row >= B) return;

  const uint64_t rowbase = (uint64_t)row * KDIM;
  const float* zrow = z_g + rowbase;
  const float* urow = u_g + rowbase;

  // Low 32 bits of a flat pointer into LDS == wave-relative LDS byte address.
  const unsigned lds_z = (unsigned)(uintptr_t)&s_z[wave][0];
  const unsigned lds_u = (unsigned)(uintptr_t)&s_u[wave][0];

  // ---- Stage both rows into LDS with CDNA5 async global->LDS loads --------
  // global_load_async_to_lds_b128: VDST = LDS byte addr (per lane),
  // VADDR = 32-bit byte offset, SADDR = 64-bit SGPR base (GVS mode).
#pragma unroll
  for (int c = 0; c < CHUNKS; ++c) {
    const unsigned boff = (unsigned)(c * 128 + lane * 4) * 4u;  // byte offset
    const unsigned dz = lds_z + boff;
    const unsigned du = lds_u + boff;
    asm volatile("global_load_async_to_lds_b128 %0, %1, %2"
                 :: "v"(dz), "v"(boff), "s"(zrow) : "memory");
    asm volatile("global_load_async_to_lds_b128 %0, %1, %2"
                 :: "v"(du), "v"(boff), "s"(urow) : "memory");
  }
  // Wave only consumes its own async writes -> per-wave asynccnt wait suffices.
  asm volatile("s_wait_asynccnt 0" ::: "memory");

  // ---- LDS -> registers (row stays register-resident for all iterations) --
  v4f zr[CHUNKS], ur[CHUNKS];
#pragma unroll
  for (int c = 0; c < CHUNKS; ++c) {
    const int e = c * 128 + lane * 4;
    zr[c] = *(const v4f*)&s_z[wave][e];
    ur[c] = *(const v4f*)&s_u[wave][e];
  }

  // ---- Bracket: f(lo)=sum(u)>=1, f(hi)=0 ----------------------------------
  float hi_l = -__builtin_inff(), lo_l = __builtin_inff();
#pragma unroll
  for (int c = 0; c < CHUNKS; ++c) {
#pragma unroll
    for (int j = 0; j < 4; ++j) {
      hi_l = fmaxf(hi_l, zr[c][j]);
      lo_l = fminf(lo_l, zr[c][j] - ur[c][j]);
    }
  }
  float lo = wred_min(lo_l);
  float hi = wred_max(hi_l);
  float tau = 0.5f * (lo + hi);

  // ---- Phase 1: cheap bisection (sub + clamp + add per element) -----------
#pragma unroll 1
  for (int it = 0; it < BISECT_ITERS; ++it) {
    float acc = 0.0f;
#pragma unroll
    for (int c = 0; c < CHUNKS; ++c) {
#pragma unroll
      for (int j = 0; j < 4; ++j) {
        const float t = zr[c][j] - tau;
        acc += fminf(fmaxf(t, 0.0f), ur[c][j]);
      }
    }
    const float f = wred_sum(acc);
    if (f > 1.0f) lo = tau; else hi = tau;
    tau = 0.5f * (lo + hi);
  }

  // ---- Phase 2: safeguarded Newton on the piecewise-linear segment --------
  // Segment stats at tau: w = sum_active(z) + sum_saturated(u), cnt = #active.
  // Then f(tau) = w - cnt*tau and the segment-exact root is (w-1)/cnt.
#pragma unroll 1
  for (int it = 0; it < NEWTON_ITERS; ++it) {
    float wl = 0.0f, cntl = 0.0f;
#pragma unroll
    for (int c = 0; c < CHUNKS; ++c) {
#pragma unroll
      for (int j = 0; j < 4; ++j) {
        const float zz = zr[c][j], uu = ur[c][j];
        const float t = zz - tau;
        const bool sat = (t >= uu);
        const bool act = (t > 0.0f) && !sat;   // 0 < t < u
        const float contrib = act ? zz : (sat ? uu : 0.0f);
        wl   += contrib;
        cntl += act ? 1.0f : 0.0f;
      }
    }
    const float w   = wred_sum(wl);
    const float cnt = wred_sum(cntl);
    const float f   = w - cnt * tau;
    if (f > 1.0f) lo = tau; else hi = tau;
    const float cand = (w - 1.0f) / fmaxf(cnt, 1.0f);
    tau = (cnt > 0.5f && cand > lo && cand < hi) ? cand : 0.5f * (lo + hi);
  }

  // ---- Emit p, regions, val ----------------------------------------------
  float vall = 0.0f;
#pragma unroll
  for (int c = 0; c < CHUNKS; ++c) {
    v4f pv;
    v4i rv;
#pragma unroll
    for (int j = 0; j < 4; ++j) {
      const float zz = zr[c][j], uu = ur[c][j];
      const float t = zz - tau;
      const float p = fminf(fmaxf(t, 0.0f), uu);
      pv[j] = p;
      rv[j] = (p <= 0.0f) ? 0 : ((p >= uu) ? 2 : 1);
      const float d = p - zz;
      vall = fmaf(d, d, vall);
    }
    const uint64_t e = rowbase + (uint64_t)(c * 128 + lane * 4);
    *(v4f*)&p_out[e] = pv;
    *(v4i*)&r_out[e] = rv;
  }
  const float val = 0.5f * wred_sum(vall);
  if (lane == 0) {
    tau_out[row] = tau;
    val_out[row] = val;
  }
}

extern "C" void kernel_launch(void* const* d_in, const int* in_sizes, int n_in,
                              void* d_out, int out_size, void* d_ws, size_t ws_size,
                              hipStream_t stream) {
  (void)n_in; (void)out_size; (void)d_ws; (void)ws_size;
  const float* z = (const float*)d_in[0];
  const float* u = (const float*)d_in[1];
  const int B = in_sizes[0] / KDIM;

  float* out = (float*)d_out;
  const size_t BK = (size_t)B * KDIM;
  float* p_out   = out;
  int*   r_out   = (int*)(out + BK);
  float* tau_out = out + 2 * BK;
  float* val_out = out + 2 * BK + B;

  const int blocks = (B + WAVES_PER_BLOCK - 1) / WAVES_PER_BLOCK;
  hipLaunchKernelGGL(ConstrainedSparsemax_kernel, dim3(blocks),
                     dim3(WAVES_PER_BLOCK * LANES), 0, stream,
                     z, u, p_out, r_out, tau_out, val_out, B);
}